// UpSampleOffset_11390253269668
// MI455X (gfx1250) — compile-verified
//
#include <hip/hip_runtime.h>

#define NC_IN   96
#define NC_UP   32
#define NC_OUT  64
#define KNBR    27
#define EPSV    1e-5f

typedef float v2f __attribute__((ext_vector_type(2)));
typedef float v4f __attribute__((ext_vector_type(4)));
typedef float v8f __attribute__((ext_vector_type(8)));

// ---------------------------------------------------------------- utilities

// Zeros the BN-stat accumulators AND the 96-float zero-row used for invalid
// neighbors. Runs every launch (workspace is not re-poisoned but also not
// preserved-by-contract, so we re-zero deterministically).
__global__ void zero_stats_kernel(float* p, int n) {
    int t = blockIdx.x * blockDim.x + threadIdx.x;
    if (t < n) p[t] = 0.f;
}

// scale = g * rsqrt(var + eps); shift = b - mean*scale
__global__ void finalize_bn_kernel(const float* __restrict__ sum,
                                   const float* __restrict__ sumsq,
                                   const float* __restrict__ g,
                                   const float* __restrict__ b,
                                   float* __restrict__ scale,
                                   float* __restrict__ shift,
                                   int C, float invN) {
    int c = threadIdx.x;
    if (c < C) {
        float mean = sum[c] * invN;
        float var  = sumsq[c] * invN - mean * mean;
        float sc   = g[c] * rsqrtf(var + EPSV);
        scale[c] = sc;
        shift[c] = b[c] - mean * sc;
    }
}

__global__ void apply_bn_relu_kernel(const float* __restrict__ yraw,
                                     const float* __restrict__ scale,
                                     const float* __restrict__ shift,
                                     float* __restrict__ out, int total) {
    int t = blockIdx.x * blockDim.x + threadIdx.x;
    if (t >= total) return;
    int c = t & (NC_OUT - 1);
    out[t] = fmaxf(fmaf(yraw[t], scale[c], shift[c]), 0.f);
}

// ---------------------------------------------------------------- weight repack
// Rearranges W[27][CIN][64] into the exact per-lane WMMA B-fragment order:
//   Brep[((k*(CIN/4)+g)*32 + lane)*8 + e]
// where for lane (hf=lane>>4, m=lane&15) and e (j=e>>1, which=e&1):
//   value = W[k][4g + 2*hf + which][16j + m]
// so in the conv hot loop each lane reads its 8 B-values (all 4 output
// chunks of one K-group) with two quad-aligned b128 loads at immediate
// offsets. ~1.1 MB total, one-time per launch.
template <int CIN>
__global__ void repack_weights_kernel(const float* __restrict__ W,
                                      float* __restrict__ Brep) {
    int t = blockIdx.x * blockDim.x + threadIdx.x;
    if (t >= KNBR * CIN * NC_OUT) return;
    int e     = t & 7;
    int lane  = (t >> 3) & 31;
    int kg    = t >> 8;                 // k*(CIN/4) + g
    int g     = kg % (CIN / 4);
    int k     = kg / (CIN / 4);
    int j     = e >> 1;
    int which = e & 1;
    int hf    = lane >> 4;
    int m     = lane & 15;
    int c     = 4 * g + 2 * hf + which;
    Brep[t] = W[((size_t)k * CIN + c) * NC_OUT + 16 * j + m];
}

// ---------------------------------------------------------------- upsample
// up_raw[i, co] = sum_c x1[src[i], c] * Wup[off[i], c, co]; also BN stats.
// Tiny stage (~1.2 GFLOP): 8 threads/point, each computing 4 channels with
// b128 weight loads.
__global__ __launch_bounds__(256) void upsample_kernel(
        const float* __restrict__ x1, const float* __restrict__ Wup,
        const int* __restrict__ src_idx, const int* __restrict__ off_id,
        float* __restrict__ up_raw,
        float* __restrict__ sum, float* __restrict__ sumsq, int n) {
    int t = blockIdx.x * blockDim.x + threadIdx.x;
    int q  = (t & 7) * 4;               // channel quad base
    int i0 = t >> 3;
    int stride = (gridDim.x * blockDim.x) >> 3;
    v4f s = {}, sq = {};
    for (int i = i0; i < n; i += stride) {
        int src = src_idx[i];
        int off = off_id[i];
        const float* xr = x1 + (size_t)src * NC_IN;
        const float* w  = Wup + (size_t)off * NC_IN * NC_UP + q;
        v4f acc = {};
        #pragma unroll 4
        for (int c = 0; c < NC_IN; ++c) {
            v4f wv = *(const v4f*)(w + (size_t)c * NC_UP);
            float xv = xr[c];
            acc += wv * xv;
        }
        *(v4f*)(up_raw + (size_t)i * NC_UP + q) = acc;
        s += acc;
        sq += acc * acc;
    }
    #pragma unroll
    for (int e = 0; e < 4; ++e) {
        atomicAdd(&sum[q + e], s[e]);
        atomicAdd(&sumsq[q + e], sq[e]);
    }
}

// x[i, 0:32] = relu(bn(up_raw)); x[i,32:64] = x2; x[i,64:96] = x3
__global__ void build_x_kernel(const float* __restrict__ up_raw,
                               const float* __restrict__ x2,
                               const float* __restrict__ x3,
                               const float* __restrict__ scale,
                               const float* __restrict__ shift,
                               float* __restrict__ x, int n) {
    int t = blockIdx.x * blockDim.x + threadIdx.x;
    if (t >= n * NC_IN) return;
    int i = t / NC_IN;
    int c = t - i * NC_IN;
    float v;
    if (c < NC_UP) {
        float u = up_raw[(size_t)i * NC_UP + c];
        v = fmaxf(fmaf(u, scale[c], shift[c]), 0.f);
    } else if (c < 2 * NC_UP) {
        v = x2[(size_t)i * NC_UP + (c - NC_UP)];
    } else {
        v = x3[(size_t)i * NC_UP + (c - 2 * NC_UP)];
    }
    x[t] = v;
}

// ---------------------------------------------------------------- conv (WMMA)
// One wave32 per 16-point tile; D = 16 pts x 64 out-ch in 4 f32 accumulators.
// V_WMMA_F32_16X16X4_F32 over K = 27 neighbors x CIN channels.
// A fragment (16x4, 2 VGPRs): lane L = row (L&15), K-half (L>>4) -> one
//   contiguous b64 from the gathered feature row. Invalid neighbors read a
//   zeroed row (no per-element cndmask, EXEC stays all-ones as WMMA needs).
// B fragment: two b128 loads/K-group from the repacked stream; the even-pair
//   sub-vectors are the WMMA B operands directly (even-VGPR aligned).
template <int CIN>
__global__ __launch_bounds__(256) void subm_conv_wmma_kernel(
        const float* __restrict__ feats,   // n_tiles*16 x CIN
        const float* __restrict__ Brep,    // repacked weights
        const int*   __restrict__ nbr,     // n x 27
        const float* __restrict__ zrow,    // CIN zeros
        float* __restrict__ out,           // n x 64
        float* __restrict__ sum, float* __restrict__ sumsq,
        int n_tiles) {
    const int wave = threadIdx.x >> 5;
    const int lane = threadIdx.x & 31;
    const int tile = blockIdx.x * (blockDim.x >> 5) + wave;
    if (tile >= n_tiles) return;
    const int hf = lane >> 4;       // K-half / M-half selector
    const int m  = lane & 15;       // row (A) / column (B,D)
    const int row = tile * 16 + m;

    v8f acc[4];
    #pragma unroll
    for (int j = 0; j < 4; ++j) acc[j] = v8f{};

    const float* __restrict__ bl = Brep + (size_t)lane * 8;

    #pragma unroll 1
    for (int k = 0; k < KNBR; ++k) {
        int idx = nbr[(size_t)row * KNBR + k];
        const float* __restrict__ rp =
            (idx >= 0) ? (feats + (size_t)idx * CIN) : zrow;
        const float* __restrict__ ap = rp + 2 * hf;
        const float* __restrict__ bk = bl + (size_t)k * (CIN / 4) * 256;
        #pragma unroll
        for (int g = 0; g < CIN / 4; ++g) {
            v2f a   = *(const v2f*)(ap + 4 * g);
            v4f b01 = *(const v4f*)(bk + (size_t)g * 256);
            v4f b23 = *(const v4f*)(bk + (size_t)g * 256 + 4);
            v2f b;
            b[0] = b01[0]; b[1] = b01[1];
            acc[0] = __builtin_amdgcn_wmma_f32_16x16x4_f32(
                false, a, false, b, (short)0, acc[0], false, false);
            b[0] = b01[2]; b[1] = b01[3];
            acc[1] = __builtin_amdgcn_wmma_f32_16x16x4_f32(
                false, a, false, b, (short)0, acc[1], false, false);
            b[0] = b23[0]; b[1] = b23[1];
            acc[2] = __builtin_amdgcn_wmma_f32_16x16x4_f32(
                false, a, false, b, (short)0, acc[2], false, false);
            b[0] = b23[2]; b[1] = b23[3];
            acc[3] = __builtin_amdgcn_wmma_f32_16x16x4_f32(
                false, a, false, b, (short)0, acc[3], false, false);
        }
    }

    // D layout: VGPR r <-> row M = hf*8 + r; channel = 16*j + m.
    const size_t obase = (size_t)(tile * 16 + hf * 8) * NC_OUT + m;
    #pragma unroll
    for (int j = 0; j < 4; ++j) {
        float s = 0.f, sq = 0.f;
        #pragma unroll
        for (int r = 0; r < 8; ++r) {
            float v = acc[j][r];
            out[obase + (size_t)r * NC_OUT + 16 * j] = v;
            s += v; sq += v * v;
        }
        atomicAdd(&sum[16 * j + m], s);
        atomicAdd(&sumsq[16 * j + m], sq);
    }
}

// ---------------------------------------------------------------- launcher

extern "C" void kernel_launch(void* const* d_in, const int* in_sizes, int n_in,
                              void* d_out, int out_size, void* d_ws, size_t ws_size,
                              hipStream_t stream) {
    const float* x1   = (const float*)d_in[0];
    const float* x2   = (const float*)d_in[1];
    const float* x3   = (const float*)d_in[2];
    const float* Wup  = (const float*)d_in[3];
    const float* g_up = (const float*)d_in[4];
    const float* b_up = (const float*)d_in[5];
    const float* W1   = (const float*)d_in[6];
    const float* g1   = (const float*)d_in[7];
    const float* b1   = (const float*)d_in[8];
    const float* W2   = (const float*)d_in[9];
    const float* g2   = (const float*)d_in[10];
    const float* b2   = (const float*)d_in[11];
    const int* up_src = (const int*)d_in[12];
    const int* up_off = (const int*)d_in[13];
    const int* nbr    = (const int*)d_in[14];
    float* out = (float*)d_out;

    const int N = in_sizes[12];          // number of fine points
    const float invN = 1.f / (float)N;

    // workspace layout (floats):
    //   x (N*96) | bufA (N*64) | bufB (N*64) | stats+zrow (1024) |
    //   Brep1 (27*96*64) | Brep2 (27*64*64)
    float* xbuf  = (float*)d_ws;
    float* bufA  = xbuf + (size_t)N * NC_IN;
    float* bufB  = bufA + (size_t)N * NC_OUT;
    float* stats = bufB + (size_t)N * NC_OUT;
    float* sum_up = stats +   0; float* sq_up = stats +  32;
    float* sum1   = stats +  64; float* sq1   = stats + 128;
    float* sum2   = stats + 192; float* sq2   = stats + 256;
    float* sc_up  = stats + 320; float* sh_up = stats + 352;
    float* sc1    = stats + 384; float* sh1   = stats + 448;
    float* sc2    = stats + 512; float* sh2   = stats + 576;
    float* zrow   = stats + 640;              // 96 zeros
    float* Brep1  = stats + 1024;
    float* Brep2  = Brep1 + (size_t)KNBR * NC_IN * NC_OUT;

    // zero BN accumulators (0..319) and zero-row (640..735)
    zero_stats_kernel<<<3, 256, 0, stream>>>(stats, 736);

    // one-time weight repacks into WMMA B-fragment order
    {
        int t1 = KNBR * NC_IN * NC_OUT;
        repack_weights_kernel<NC_IN><<<(t1 + 255) / 256, 256, 0, stream>>>(W1, Brep1);
        int t2 = KNBR * NC_OUT * NC_OUT;
        repack_weights_kernel<NC_OUT><<<(t2 + 255) / 256, 256, 0, stream>>>(W2, Brep2);
    }

    // stage 1: upsample + stats, BN finalize, build concat input
    upsample_kernel<<<512, 256, 0, stream>>>(x1, Wup, up_src, up_off,
                                             bufA, sum_up, sq_up, N);
    finalize_bn_kernel<<<1, 64, 0, stream>>>(sum_up, sq_up, g_up, b_up,
                                             sc_up, sh_up, NC_UP, invN);
    {
        int total = N * NC_IN;
        build_x_kernel<<<(total + 255) / 256, 256, 0, stream>>>(
            bufA, x2, x3, sc_up, sh_up, xbuf, N);
    }

    const int n_tiles = (N + 15) / 16;       // N = 200000 -> 12500 exact
    const int cblocks = (n_tiles + 7) / 8;   // 8 waves / block

    // stage 2: conv1 (96 -> 64) with f32 WMMA, BN, ReLU
    subm_conv_wmma_kernel<NC_IN><<<cblocks, 256, 0, stream>>>(
        xbuf, Brep1, nbr, zrow, bufB, sum1, sq1, n_tiles);
    finalize_bn_kernel<<<1, 64, 0, stream>>>(sum1, sq1, g1, b1, sc1, sh1,
                                             NC_OUT, invN);
    {
        int total = N * NC_OUT;
        apply_bn_relu_kernel<<<(total + 255) / 256, 256, 0, stream>>>(
            bufB, sc1, sh1, bufA, total);
    }

    // stage 3: conv2 (64 -> 64) with f32 WMMA, BN, ReLU -> d_out
    subm_conv_wmma_kernel<NC_OUT><<<cblocks, 256, 0, stream>>>(
        bufA, Brep2, nbr, zrow, bufB, sum2, sq2, n_tiles);
    finalize_bn_kernel<<<1, 64, 0, stream>>>(sum2, sq2, g2, b2, sc2, sh2,
                                             NC_OUT, invN);
    {
        int total = N * NC_OUT;
        apply_bn_relu_kernel<<<(total + 255) / 256, 256, 0, stream>>>(
            bufB, sc2, sh2, out, total);
    }
}